// CNODEmod_69492570849403
// MI455X (gfx1250) — compile-verified
//
#include <hip/hip_runtime.h>

typedef float v2f __attribute__((ext_vector_type(2)));
typedef float v8f __attribute__((ext_vector_type(8)));

#define NC     64
#define TOBS   50
#define BATCHN 2048
#define DTF    0.05f
#define SROW   20     // padded LDS row stride (floats) for 16-wide state rows

// d_ws float offsets
#define WS_R    0       // Aop for one-Euler-step matrix: R[c][k] = (c==k) + DT*A[c][k]
#define WS_W1T  4096    // Aop for MLP layer1: W1T[j][k] = W1[k][j]
#define WS_WG   8192    // Aop for bridge: R^10
#define WS_UP   12288   // u_pred[64]
#define WS_UO   12352   // u_obs[64]
#define WS_DB   12416   // DT*Bv[64]
#define WS_BV   12480   // Bv[64]
#define WS_T1   16384
#define WS_T2   20480
#define WS_T3   24576
#define WS_PA   28672
#define WS_PB   28736

// ---------------- setup: build HiPPO matrices, Q^10, bridge vectors ----------------
__global__ void cnode_setup(const float* __restrict__ W1, float* __restrict__ ws) {
  const int c = threadIdx.x;  // 0..63
  float* R   = ws + WS_R;
  float* W1T = ws + WS_W1T;
  float* WG  = ws + WS_WG;
  float* T1  = ws + WS_T1;
  float* T2  = ws + WS_T2;
  float* T3  = ws + WS_T3;
  float* PA  = ws + WS_PA;
  float* PB  = ws + WS_PB;

  const float sqc = sqrtf(2.0f * (float)c + 1.0f);
  for (int k = 0; k < NC; ++k) {
    float sqk = sqrtf(2.0f * (float)k + 1.0f);
    float sgn = (k <= c) ? 1.0f : (((c - k) & 1) ? -1.0f : 1.0f);
    float a   = -(1.0f / 5.0f) * sqc * sqk * sgn;          // A[c][k]
    R[c * NC + k]   = ((c == k) ? 1.0f : 0.0f) + DTF * a;  // (I + DT*A^T) transposed
    W1T[c * NC + k] = W1[k * NC + c];
  }
  const float bv = sqc / 5.0f;
  ws[WS_BV + c] = bv;
  ws[WS_DB + c] = DTF * bv;
  __syncthreads();

  auto mm = [&](float* D, const float* X, const float* Y) {
    for (int k = 0; k < NC; ++k) {
      float s = 0.0f;
      for (int m = 0; m < NC; ++m) s = fmaf(X[c * NC + m], Y[m * NC + k], s);
      D[c * NC + k] = s;
    }
  };
  mm(T1, R, R);   __syncthreads();  // R^2
  mm(T2, T1, T1); __syncthreads();  // R^4
  mm(T3, T2, T2); __syncthreads();  // R^8
  mm(WG, T3, T1); __syncthreads();  // R^10

  // p_j = R^j * Bv ; u_pred = sum_j DT*(j+1)/10 * p_j ; u_obs = sum_j DT*(9-j)/10 * p_j
  PA[c] = bv;
  __syncthreads();
  float up = 0.0f, uo = 0.0f;
  float* pa = PA; float* pb = PB;
  for (int j = 0; j < 10; ++j) {
    up = fmaf(DTF * 0.1f * (float)(j + 1), pa[c], up);
    uo = fmaf(DTF * 0.1f * (float)(9 - j), pa[c], uo);
    float s = 0.0f;
    for (int k = 0; k < NC; ++k) s = fmaf(R[c * NC + k], pa[k], s);
    pb[c] = s;
    __syncthreads();
    float* t = pa; pa = pb; pb = t;
  }
  ws[WS_UP + c] = up;
  ws[WS_UO + c] = uo;
}

// ---------------- main scan kernel: one wave per 16-batch tile ----------------
__global__ __launch_bounds__(32) void cnode_scan(
    const float* __restrict__ Y, const int* __restrict__ mask,
    const float* __restrict__ b1, const float* __restrict__ W2,
    const float* __restrict__ b2, const float* __restrict__ ws,
    float* __restrict__ out) {
  __shared__ float sm[3 * NC * SROW];

  const int tid  = threadIdx.x;
  const int lid  = tid & 15;   // batch column within tile (both half-waves)
  const int hi   = tid >> 4;   // 0 = lanes 0-15, 1 = lanes 16-31
  const int tile = blockIdx.x; // 0..127
  const int bg   = tile * 16 + lid;

  const float* WA  = ws + WS_R;
  const float* W1T = ws + WS_W1T;
  const float* WG  = ws + WS_WG;
  const float* UP  = ws + WS_UP;
  const float* UO  = ws + WS_UO;
  const float* DB  = ws + WS_DB;

  // per-lane C/D-layout constants: index m = t4*16 + r + hi*8
  float b1v[32], w2v[32], dbv[32];
#pragma unroll
  for (int t4 = 0; t4 < 4; ++t4)
#pragma unroll
    for (int r = 0; r < 8; ++r) {
      int m = t4 * 16 + r + hi * 8;
      b1v[t4 * 8 + r] = b1[m];
      w2v[t4 * 8 + r] = W2[m];
      dbv[t4 * 8 + r] = DB[m];
    }
  const float b2s = b2[0];

  float* r0 = sm;
  float* r1 = sm + NC * SROW;
  float* r2 = sm + 2 * NC * SROW;

  // h0 = 0
  for (int c2 = 0; c2 < NC; ++c2) r0[c2 * SROW + lid] = 0.0f;

  v2f B[16];  // B-operand state chunks (K = 4 per chunk)

  auto loadB = [&](const float* src) {
#pragma unroll
    for (int kc = 0; kc < 16; ++kc) {
      int row = kc * 4 + hi * 2;
      v2f b;
      b.x = src[row * SROW + lid];
      b.y = src[(row + 1) * SROW + lid];
      B[kc] = b;
    }
  };
  auto loadA = [&](const float* Wm, int t4, int kc) -> v2f {
    const float* p = Wm + (t4 * 16 + lid) * NC + kc * 4 + hi * 2;
    v2f a; a.x = p[0]; a.y = p[1];
    return a;
  };

  // node(h) for state currently in B[]: z = relu(h@W1 + b1); y = z@W2 + b2
  auto mlp_y = [&]() -> float {
    float ys = 0.0f;
#pragma unroll
    for (int t4 = 0; t4 < 4; ++t4) {
      v8f acc;
#pragma unroll
      for (int r = 0; r < 8; ++r) acc[r] = b1v[t4 * 8 + r];
#pragma unroll
      for (int kc = 0; kc < 16; ++kc)
        acc = __builtin_amdgcn_wmma_f32_16x16x4_f32(
            false, loadA(W1T, t4, kc), false, B[kc], (short)0, acc, false, false);
#pragma unroll
      for (int r = 0; r < 8; ++r) {
        float z = acc[r] > 0.0f ? acc[r] : 0.0f;
        ys = fmaf(z, w2v[t4 * 8 + r], ys);
      }
    }
    ys += __shfl_xor(ys, 16, 32);  // combine the two half-wave partials
    return ys + b2s;
  };

  // one Euler step: dst = h@(I+DT*A^T) + (DT*y)*Bv   (state from B[])
  auto advance = [&](float y, float* dst) {
#pragma unroll
    for (int t4 = 0; t4 < 4; ++t4) {
      v8f acc;
#pragma unroll
      for (int r = 0; r < 8; ++r) acc[r] = 0.0f;
#pragma unroll
      for (int kc = 0; kc < 16; ++kc)
        acc = __builtin_amdgcn_wmma_f32_16x16x4_f32(
            false, loadA(WA, t4, kc), false, B[kc], (short)0, acc, false, false);
#pragma unroll
      for (int r = 0; r < 8; ++r) {
        float v = fmaf(y, dbv[t4 * 8 + r], acc[r]);
        dst[(t4 * 16 + r + hi * 8) * SROW + lid] = v;
      }
    }
  };

  // exact 10-step linear bridge: dst = h_pre@Q^10 + yp*u_pred + yo*u_obs
  auto bridge = [&](float yp, float yo, float* dst) {
#pragma unroll
    for (int t4 = 0; t4 < 4; ++t4) {
      v8f acc;
#pragma unroll
      for (int r = 0; r < 8; ++r) acc[r] = 0.0f;
#pragma unroll
      for (int kc = 0; kc < 16; ++kc)
        acc = __builtin_amdgcn_wmma_f32_16x16x4_f32(
            false, loadA(WG, t4, kc), false, B[kc], (short)0, acc, false, false);
#pragma unroll
      for (int r = 0; r < 8; ++r) {
        int m = t4 * 16 + r + hi * 8;
        float v = fmaf(yp, UP[m], fmaf(yo, UO[m], acc[r]));
        dst[m * SROW + lid] = v;
      }
    }
  };

#pragma unroll 1
  for (int t = 0; t < TOBS; ++t) {
    // phase 1: 10 ode_fun steps, r0 -> ... -> h_pre in r2
#pragma unroll 1
    for (int s = 0; s < 10; ++s) {
      float* src = (s == 0) ? r0 : ((s & 1) ? r1 : r2);
      float* dst = (s & 1) ? r2 : r1;
      loadB(src);
      float y = mlp_y();
      advance(y, dst);
    }
    // phase 2: 10 ode_fun steps from h_pre (r2) -> h_nc in r0.
    // Step 0 also yields y_pred = node(h_pre) and shares B[] with the bridge.
    float mf = 0.0f;
#pragma unroll 1
    for (int s = 0; s < 10; ++s) {
      float* src = (s == 0) ? r2 : ((s & 1) ? r1 : r0);
      float* dst = (s & 1) ? r0 : r1;
      loadB(src);
      float y = mlp_y();
      if (s == 0) {
        if (hi == 0) out[t * BATCHN + bg] = y;       // preds[t][b][0]
        float yobs = Y[bg * TOBS + t];
        mf = (float)mask[bg * TOBS + t];
        advance(y, dst);                              // first h_nc step -> r1
        bridge(y, yobs, r2);                          // h_upd -> r2 (B[] already read)
      } else {
        advance(y, dst);
      }
    }
    // blend: r0 = (1-m)*h_nc + m*h_upd   -> new current state in r0
#pragma unroll
    for (int t4 = 0; t4 < 4; ++t4)
#pragma unroll
      for (int r = 0; r < 8; ++r) {
        int idx = (t4 * 16 + r + hi * 8) * SROW + lid;
        float hn = r0[idx];
        float hu = r2[idx];
        r0[idx] = fmaf(mf, hu - hn, hn);
      }
  }
}

extern "C" void kernel_launch(void* const* d_in, const int* in_sizes, int n_in,
                              void* d_out, int out_size, void* d_ws, size_t ws_size,
                              hipStream_t stream) {
  (void)in_sizes; (void)n_in; (void)out_size; (void)ws_size;
  const float* Y    = (const float*)d_in[1];
  const int*   mask = (const int*)d_in[2];
  const float* W1   = (const float*)d_in[3];
  const float* b1   = (const float*)d_in[4];
  const float* W2   = (const float*)d_in[5];
  const float* b2   = (const float*)d_in[6];
  float* ws  = (float*)d_ws;
  float* out = (float*)d_out;

  cnode_setup<<<1, 64, 0, stream>>>(W1, ws);
  cnode_scan<<<BATCHN / 16, 32, 0, stream>>>(Y, mask, b1, W2, b2, ws, out);
}